// Attention_26568667693839
// MI455X (gfx1250) — compile-verified
//
#include <hip/hip_runtime.h>

typedef __attribute__((ext_vector_type(2))) float v2f;
typedef __attribute__((ext_vector_type(8))) float v8f;

// D = A(16x4) * B(4x16) + C, fp32 WMMA
__device__ __forceinline__ v8f wmma4(v2f a, v2f b, v8f c) {
  return __builtin_amdgcn_wmma_f32_16x16x4_f32(false, a, false, b, (short)0, c,
                                               false, false);
}

#define NSEQ   1028
#define CDIM   1024
#define HEADS  16
#define HD     64
#define BATCH  4
#define MROWS  (BATCH * NSEQ)      // 4112
#define QT_N   65                  // ceil(1028/16)

// ---------------------------------------------------------------------------
// Generic C = A(MxK) * B(NxK)^T, row-major; M,K multiples of 16, N of 64,
// K multiple of 32. One wave computes a 16x64 strip (1 M-tile x 4 N-tiles).
// Double-buffered 16-wide K slabs so loads of slab n+1 overlap WMMAs of slab n.
// ---------------------------------------------------------------------------
__global__ __launch_bounds__(256) void gemm_nt_wmma(
    const float* __restrict__ A, const float* __restrict__ Bm,
    float* __restrict__ C, int M, int N, int K, int Ngrp) {
  int wave = (blockIdx.x * blockDim.x + threadIdx.x) >> 5;
  int lane = threadIdx.x & 31;
  int Mt = M >> 4;
  if (wave >= Mt * Ngrp) return;
  int mt = wave / Ngrp, ng = wave % Ngrp;
  int mBase = mt << 4, nBase = ng << 6;
  int lm = lane & 15, lh = lane >> 4;

  const float* ap  = A  + (size_t)(mBase + lm) * K + 2 * lh;
  const float* bp0 = Bm + (size_t)(nBase + lm) * K + 2 * lh;
  const float* bp1 = bp0 + 16 * (size_t)K;
  const float* bp2 = bp0 + 32 * (size_t)K;
  const float* bp3 = bp0 + 48 * (size_t)K;

  v8f acc0 = {}, acc1 = {}, acc2 = {}, acc3 = {};
  v2f a0[4], a1[4], b0[4][4], b1[4][4];

#define LOAD_SLAB(aa, bb, kbase)                                               \
  _Pragma("unroll") for (int c = 0; c < 4; c++) {                              \
    int kc = (kbase) + 4 * c;                                                  \
    aa[c]    = *(const v2f*)(ap + kc);                                         \
    bb[0][c] = *(const v2f*)(bp0 + kc);                                        \
    bb[1][c] = *(const v2f*)(bp1 + kc);                                        \
    bb[2][c] = *(const v2f*)(bp2 + kc);                                        \
    bb[3][c] = *(const v2f*)(bp3 + kc);                                        \
  }

#define COMP_SLAB(aa, bb)                                                      \
  _Pragma("unroll") for (int c = 0; c < 4; c++) {                              \
    acc0 = wmma4(aa[c], bb[0][c], acc0);                                       \
    acc1 = wmma4(aa[c], bb[1][c], acc1);                                       \
    acc2 = wmma4(aa[c], bb[2][c], acc2);                                       \
    acc3 = wmma4(aa[c], bb[3][c], acc3);                                       \
  }

  LOAD_SLAB(a0, b0, 0)
  for (int k0 = 0; k0 < K; k0 += 32) {
    LOAD_SLAB(a1, b1, k0 + 16)      // prefetch next slab
    COMP_SLAB(a0, b0)
    if (k0 + 32 < K) { LOAD_SLAB(a0, b0, k0 + 32) }
    COMP_SLAB(a1, b1)
  }
#undef LOAD_SLAB
#undef COMP_SLAB

  // C-layout: lane holds col n = lm; VGPR r holds row m = r + 8*lh
  float* crow = C + (size_t)(mBase + 8 * lh) * N + nBase + lm;
#pragma unroll
  for (int r = 0; r < 8; r++) {
    float* cp = crow + (size_t)r * N;
    cp[0]  = acc0[r];
    cp[16] = acc1[r];
    cp[32] = acc2[r];
    cp[48] = acc3[r];
  }
}

// ---------------------------------------------------------------------------
// RMSNorm + RoPE applied in place to q,k slices of the qkv buffer.
// One wave per (b, n, h) row; lane holds elements d=lane and d=lane+32.
// ---------------------------------------------------------------------------
__global__ __launch_bounds__(256) void normrope_kernel(
    float* __restrict__ qkv, const float* __restrict__ cosb,
    const float* __restrict__ sinb, const float* __restrict__ qw,
    const float* __restrict__ kw) {
  int wave = (blockIdx.x * blockDim.x + threadIdx.x) >> 5;
  int lane = threadIdx.x & 31;
  if (wave >= MROWS * HEADS) return;
  int h = wave & (HEADS - 1);
  int bn = wave >> 4;              // b*N + n
  int n = bn % NSEQ;

  float* qp = qkv + (size_t)bn * (3 * CDIM) + h * HD;
  float* kp = qp + CDIM;
  int d0 = lane, d1 = lane + 32;
  float q0 = qp[d0], q1 = qp[d1];
  float k0 = kp[d0], k1 = kp[d1];

  float sq = q0 * q0 + q1 * q1;
  float sk = k0 * k0 + k1 * k1;
#pragma unroll
  for (int m = 1; m < 32; m <<= 1) {
    sq += __shfl_xor(sq, m, 32);
    sk += __shfl_xor(sk, m, 32);
  }
  float rq = rsqrtf(sq * (1.0f / HD) + 1e-6f);
  float rk = rsqrtf(sk * (1.0f / HD) + 1e-6f);
  q0 *= rq * qw[d0]; q1 *= rq * qw[d1];
  k0 *= rk * kw[d0]; k1 *= rk * kw[d1];

  float c0 = cosb[n * HD + d0], c1 = cosb[n * HD + d1];
  float s0 = sinb[n * HD + d0], s1 = sinb[n * HD + d1];
  // rotate_half: out[d<32] = x[d]*c - x[d+32]*s ; out[d>=32] = x[d]*c + x[d-32]*s
  qp[d0] = q0 * c0 - q1 * s0;
  qp[d1] = q1 * c1 + q0 * s1;
  kp[d0] = k0 * c0 - k1 * s0;
  kp[d1] = k1 * c1 + k0 * s1;
}

// ---------------------------------------------------------------------------
// Flash attention, one wave per (b, h, 16-query tile).
// Computes S^T = K*Q^T so each lane owns one query column -> softmax needs only
// in-lane reductions + one shfl_xor(16). Then O^T = V^T * P^T.
// All K and V loads of a key tile are issued up front so V-load latency hides
// under the softmax VALU work.
// ---------------------------------------------------------------------------
__global__ __launch_bounds__(256) void flash_kernel(
    const float* __restrict__ qkv, float* __restrict__ attn,
    const int* __restrict__ winp) {
  int wave = (blockIdx.x * blockDim.x + threadIdx.x) >> 5;
  int lane = threadIdx.x & 31;
  if (wave >= BATCH * HEADS * QT_N) return;
  int lm = lane & 15, lh = lane >> 4;
  int bh = wave / QT_N, qt = wave % QT_N;
  int b = bh >> 4, h = bh & (HEADS - 1);
  int w = *winp;

  int qBase = qt << 4;
  int qi = qBase + lm;                       // this lane's query (both halves)
  int qrow = qi < NSEQ ? qi : NSEQ - 1;
  const float* qp = qkv + (size_t)(b * NSEQ + qrow) * (3 * CDIM) + h * HD + 2 * lh;

  // Q tile as B-operand (feature k-dim): 16 chunks of v2f
  v2f qb[16];
#pragma unroll
  for (int c = 0; c < 16; c++) qb[c] = *(const v2f*)(qp + 4 * c);

  bool qs = qi < 4;
  int qr = (qi - 4) >> 5, qc = (qi - 4) & 31;   // grid pos (gs = 32)

  float m_run = -1e30f, l_run = 0.0f;
  v8f o0 = {}, o1 = {}, o2 = {}, o3 = {};
  const float* vbase = qkv + (size_t)(b * NSEQ) * (3 * CDIM) + 2 * CDIM + h * HD;

  for (int kt = 0; kt < QT_N; kt++) {
    int kBase = kt << 4;
    int kr = kBase + lm;
    int krow = kr < NSEQ ? kr : NSEQ - 1;
    const float* kp =
        qkv + (size_t)(b * NSEQ + krow) * (3 * CDIM) + CDIM + h * HD + 2 * lh;

    // ---- issue ALL loads for this key tile first ----
    v2f ka[16];
#pragma unroll
    for (int c = 0; c < 16; c++) ka[c] = *(const v2f*)(kp + 4 * c);

    float vax[4][4], vay[4][4];                // V^T operand, loaded early
#pragma unroll
    for (int c = 0; c < 4; c++) {
      int kk = kBase + 4 * c + 2 * lh;
      int vr0 = kk < NSEQ ? kk : NSEQ - 1;
      int vr1 = kk + 1 < NSEQ ? kk + 1 : NSEQ - 1;
      const float* v0 = vbase + (size_t)vr0 * (3 * CDIM) + lm;
      const float* v1 = vbase + (size_t)vr1 * (3 * CDIM) + lm;
#pragma unroll
      for (int t = 0; t < 4; t++) { vax[c][t] = v0[t * 16]; vay[c][t] = v1[t * 16]; }
    }

    // S^T tile = K (A-operand) * Q^T (B-operand)
    v8f st = {};
#pragma unroll
    for (int c = 0; c < 16; c++) st = wmma4(ka[c], qb[c], st);

    // scale + mask + partial softmax (element: key kBase+r+8*lh, query qi)
    float p[8];
    float mx = -INFINITY;
#pragma unroll
    for (int r = 0; r < 8; r++) {
      int kj = kBase + r + 8 * lh;
      float s = st[r] * 0.125f;               // 1/sqrt(64)
      bool ok;
      if (kj >= NSEQ) ok = false;
      else if (qs || kj < 4) ok = true;
      else {
        int krr = (kj - 4) >> 5, kcc = (kj - 4) & 31;
        int dr = qr - krr; dr = dr < 0 ? -dr : dr;
        int dc = qc - kcc; dc = dc < 0 ? -dc : dc;
        ok = (dr <= w) && (dc <= w);
      }
      s = ok ? s : -INFINITY;
      p[r] = s;
      mx = fmaxf(mx, s);
    }
    mx = fmaxf(mx, __shfl_xor(mx, 16, 32));   // combine both key-halves
    float m_new = fmaxf(m_run, mx);
    float f = __expf(m_run - m_new);
    float rs = 0.0f;
#pragma unroll
    for (int r = 0; r < 8; r++) { p[r] = __expf(p[r] - m_new); rs += p[r]; }
    rs += __shfl_xor(rs, 16, 32);
    l_run = l_run * f + rs;
    m_run = m_new;
#pragma unroll
    for (int r = 0; r < 8; r++) { o0[r] *= f; o1[r] *= f; o2[r] *= f; o3[r] *= f; }

    // Broadcast probs: pf[kk] = P^T[key kk][query lm] for all 32 lanes
    float pf[16];
#pragma unroll
    for (int r = 0; r < 8; r++) {
      pf[r]     = __shfl(p[r], lm, 32);
      pf[r + 8] = __shfl(p[r], lm + 16, 32);
    }

    // O^T += V^T (A-operand) * P^T (B-operand)
#pragma unroll
    for (int c = 0; c < 4; c++) {
      v2f pb;
      pb.x = lh ? pf[4 * c + 2] : pf[4 * c];
      pb.y = lh ? pf[4 * c + 3] : pf[4 * c + 1];
      v2f va;
      va.x = vax[c][0]; va.y = vay[c][0]; o0 = wmma4(va, pb, o0);
      va.x = vax[c][1]; va.y = vay[c][1]; o1 = wmma4(va, pb, o1);
      va.x = vax[c][2]; va.y = vay[c][2]; o2 = wmma4(va, pb, o2);
      va.x = vax[c][3]; va.y = vay[c][3]; o3 = wmma4(va, pb, o3);
    }
  }

  if (qi < NSEQ) {
    float inv = 1.0f / l_run;
    float* op = attn + (size_t)(b * NSEQ + qi) * CDIM + h * HD;
#pragma unroll
    for (int r = 0; r < 8; r++) {
      int d = r + 8 * lh;
      op[d]      = o0[r] * inv;
      op[16 + d] = o1[r] * inv;
      op[32 + d] = o2[r] * inv;
      op[48 + d] = o3[r] * inv;
    }
  }
}

// ---------------------------------------------------------------------------
extern "C" void kernel_launch(void* const* d_in, const int* in_sizes, int n_in,
                              void* d_out, int out_size, void* d_ws,
                              size_t ws_size, hipStream_t stream) {
  const float* hidden = (const float*)d_in[0];
  const float* cosb   = (const float*)d_in[1];
  const float* sinb   = (const float*)d_in[2];
  const float* qkv_w  = (const float*)d_in[3];
  const float* out_w  = (const float*)d_in[4];
  const float* nqw    = (const float*)d_in[5];
  const float* nkw    = (const float*)d_in[6];
  const int*   winp   = (const int*)d_in[7];
  float* out = (float*)d_out;

  float* qkv  = (float*)d_ws;                               // 4112*3072 floats
  float* attn = qkv + (size_t)MROWS * 3 * CDIM;             // 4112*1024 floats

  // 1) qkv = hidden * qkv_w^T : M=4112, N=3072, K=1024; 48 N-groups of 64
  {
    int waves = (MROWS / 16) * (3 * CDIM / 64);             // 257*48 = 12336
    gemm_nt_wmma<<<waves / 8, 256, 0, stream>>>(hidden, qkv_w, qkv,
                                                MROWS, 3 * CDIM, CDIM,
                                                3 * CDIM / 64);
  }
  // 2) RMSNorm + RoPE in place on q,k
  {
    int waves = MROWS * HEADS;                              // 65792
    normrope_kernel<<<waves / 8, 256, 0, stream>>>(qkv, cosb, sinb, nqw, nkw);
  }
  // 3) masked flash attention
  {
    int waves = BATCH * HEADS * QT_N;                       // 4160
    flash_kernel<<<waves / 8, 256, 0, stream>>>(qkv, attn, winp);
  }
  // 4) out = attn * out_w^T : M=4112, N=1024, K=1024; 16 N-groups of 64
  {
    int waves = (MROWS / 16) * (CDIM / 64);                 // 4112
    gemm_nt_wmma<<<waves / 8, 256, 0, stream>>>(attn, out_w, out,
                                                MROWS, CDIM, CDIM, CDIM / 64);
  }
}